// SoftAggCV28Exact_78726750535843
// MI455X (gfx1250) — compile-verified
//
#include <hip/hip_runtime.h>
#include <hip/hip_bf16.h>

// Problem constants (match the reference).
#define NN 262144   // tokens
#define CC 256      // channels
#define GG 4096     // segments

typedef __attribute__((ext_vector_type(2))) float v2f;
typedef __attribute__((ext_vector_type(8))) float v8f;

// D(16x16,f32) = A(16x4,f32) x B(4x16,f32) + C   -> v_wmma_f32_16x16x4_f32
// Per-lane fragments (wave32): half = lane>>4, lo = lane&15
//   A: a.x = A[m=lo][k=2*half],  a.y = A[lo][2*half+1]
//   B: b.x = B[k=2*half][n=lo],  b.y = B[2*half+1][lo]
//   C/D: c[r] = D[m = r + 8*half][n = lo]
__device__ __forceinline__ v8f wmma_f32_k4(v2f a, v2f b, v8f c) {
  return __builtin_amdgcn_wmma_f32_16x16x4_f32(
      /*neg_a=*/false, a, /*neg_b=*/false, b,
      /*c_mod=*/(short)0, c, /*reuse_a=*/false, /*reuse_b=*/false);
}

// CDNA5 async global->LDS copy (16B per lane), tracked by ASYNCcnt.
// Generic pointers to __shared__ have the LDS byte offset in their low 32 bits.
__device__ __forceinline__ void async_copy16(const float* gptr, float* lptr) {
  const unsigned lds = (unsigned)(unsigned long long)(uintptr_t)lptr;
  const unsigned long long ga = (unsigned long long)(uintptr_t)gptr;
  asm volatile("global_load_async_to_lds_b128 %0, %1, off"
               :: "v"(lds), "v"(ga) : "memory");
}
__device__ __forceinline__ void async_wait0() {
  asm volatile("s_wait_asynccnt 0x0" ::: "memory");
}

__device__ __forceinline__ float clip50(float v) {
  return fminf(fmaxf(v, -50.f), 50.f);
}

// ---------------------------------------------------------------- utilities
__global__ __launch_bounds__(256) void zero_ws_kernel(float4* __restrict__ p, int n4) {
  int i = blockIdx.x * 256 + threadIdx.x;
  if (i < n4) p[i] = make_float4(0.f, 0.f, 0.f, 0.f);
}

// Pack W (row-major [n][k]) into paired-row transposed form:
//   WP[k/2 * CC + n] = { W[n][k], W[n][k+1] }  == { W^T[k][n], W^T[k+1][n] }
// so a WMMA B-fragment is one aligned 8B load. Tiny (3 x 256KB), once per launch.
__global__ __launch_bounds__(128) void pack_w_kernel(
    const float* __restrict__ Wf, const float* __restrict__ Wg,
    const float* __restrict__ Wh, v2f* __restrict__ WfP,
    v2f* __restrict__ WgP, v2f* __restrict__ WhP) {
  const float* src;
  v2f* dst;
  if (blockIdx.y == 0)      { src = Wf; dst = WfP; }
  else if (blockIdx.y == 1) { src = Wg; dst = WgP; }
  else                      { src = Wh; dst = WhP; }
  const int n = blockIdx.x, k2 = threadIdx.x;       // k2 in [0,128)
  v2f v;
  v.x = src[n * CC + 2 * k2];
  v.y = src[n * CC + 2 * k2 + 1];
  dst[k2 * CC + n] = v;
}

// --------------------------------------------------- pass 1: denom = seg-sum(exp)
// One block = one 32-row strip of x (async-staged to LDS); 8 waves; wave w owns
// col tiles {32w, 32w+16} and BOTH 16-row halves (B fragments shared across M).
__global__ __launch_bounds__(256) void pass1_denom_kernel(
    const float* __restrict__ x, const v2f* __restrict__ WgP,
    const float* __restrict__ bg, const int* __restrict__ ix,
    float* __restrict__ denom) {
  __shared__ __align__(16) float sx[32][260];  // pad: rows land on distinct banks
  __shared__ int sseg[32];
  const int tid = threadIdx.x;
  const size_t rowbase = (size_t)blockIdx.x * 32;
  const float* xs = x + rowbase * CC;
#pragma unroll
  for (int it = 0; it < 8; ++it) {                   // 8 async b128 per thread
    const int i = tid + it * 256;
    const int r = i >> 6, c4 = (i & 63) << 2;
    async_copy16(xs + r * CC + c4, &sx[r][c4]);
  }
  if (tid < 32) sseg[tid] = ix[rowbase + tid];
  async_wait0();
  __syncthreads();

  const int wave = tid >> 5, lane = tid & 31;
  const int half = lane >> 4, lo = lane & 15;
  const int t0 = wave * 32, t1 = t0 + 16;
  v8f g00 = {}, g01 = {}, g10 = {}, g11 = {};        // [rtile][ctile]
  for (int k0 = 0; k0 < CC; k0 += 4) {
    const int kk = k0 + 2 * half;
    const v2f aLo = *(const v2f*)&sx[lo][kk];
    const v2f aHi = *(const v2f*)&sx[16 + lo][kk];
    const int kp = (k0 >> 1) + half;                 // paired-row index
    const v2f b0 = WgP[kp * CC + t0 + lo];
    const v2f b1 = WgP[kp * CC + t1 + lo];
    g00 = wmma_f32_k4(aLo, b0, g00);
    g01 = wmma_f32_k4(aLo, b1, g01);
    g10 = wmma_f32_k4(aHi, b0, g10);
    g11 = wmma_f32_k4(aHi, b1, g11);
  }
  const float bg0 = bg[t0 + lo], bg1 = bg[t1 + lo];
#pragma unroll
  for (int r = 0; r < 8; ++r) {
    const int mLo = r + 8 * half, mHi = mLo + 16;
    const size_t sLo = (size_t)sseg[mLo] * CC;
    const size_t sHi = (size_t)sseg[mHi] * CC;
    unsafeAtomicAdd(&denom[sLo + t0 + lo], __expf(clip50(g00[r] + bg0)));
    unsafeAtomicAdd(&denom[sLo + t1 + lo], __expf(clip50(g01[r] + bg1)));
    unsafeAtomicAdd(&denom[sHi + t0 + lo], __expf(clip50(g10[r] + bg0)));
    unsafeAtomicAdd(&denom[sHi + t1 + lo], __expf(clip50(g11[r] + bg1)));
  }
}

// ------------------------------- pass 2: numer = seg-sum(fx * exp/denom[seg])
__global__ __launch_bounds__(256) void pass2_numer_kernel(
    const float* __restrict__ x, const v2f* __restrict__ WfP,
    const float* __restrict__ bf, const v2f* __restrict__ WgP,
    const float* __restrict__ bg, const int* __restrict__ ix,
    const float* __restrict__ denom, float* __restrict__ numer) {
  __shared__ __align__(16) float sx[32][260];
  __shared__ int sseg[32];
  const int tid = threadIdx.x;
  const size_t rowbase = (size_t)blockIdx.x * 32;
  const float* xs = x + rowbase * CC;
#pragma unroll
  for (int it = 0; it < 8; ++it) {
    const int i = tid + it * 256;
    const int r = i >> 6, c4 = (i & 63) << 2;
    async_copy16(xs + r * CC + c4, &sx[r][c4]);
  }
  if (tid < 32) sseg[tid] = ix[rowbase + tid];
  async_wait0();
  __syncthreads();

  const int wave = tid >> 5, lane = tid & 31;
  const int half = lane >> 4, lo = lane & 15;
  const int t0 = wave * 32, t1 = t0 + 16;
  v8f f00 = {}, f01 = {}, f10 = {}, f11 = {};
  v8f g00 = {}, g01 = {}, g10 = {}, g11 = {};
  for (int k0 = 0; k0 < CC; k0 += 4) {
    const int kk = k0 + 2 * half;
    const v2f aLo = *(const v2f*)&sx[lo][kk];
    const v2f aHi = *(const v2f*)&sx[16 + lo][kk];
    const int kp = (k0 >> 1) + half;
    const v2f bF0 = WfP[kp * CC + t0 + lo];
    const v2f bF1 = WfP[kp * CC + t1 + lo];
    const v2f bG0 = WgP[kp * CC + t0 + lo];
    const v2f bG1 = WgP[kp * CC + t1 + lo];
    f00 = wmma_f32_k4(aLo, bF0, f00);
    f01 = wmma_f32_k4(aLo, bF1, f01);
    f10 = wmma_f32_k4(aHi, bF0, f10);
    f11 = wmma_f32_k4(aHi, bF1, f11);
    g00 = wmma_f32_k4(aLo, bG0, g00);
    g01 = wmma_f32_k4(aLo, bG1, g01);
    g10 = wmma_f32_k4(aHi, bG0, g10);
    g11 = wmma_f32_k4(aHi, bG1, g11);
  }
  const float bf0 = bf[t0 + lo], bf1 = bf[t1 + lo];
  const float bg0 = bg[t0 + lo], bg1 = bg[t1 + lo];
#pragma unroll
  for (int r = 0; r < 8; ++r) {
    const int mLo = r + 8 * half, mHi = mLo + 16;
    const size_t sLo = (size_t)sseg[mLo] * CC;
    const size_t sHi = (size_t)sseg[mHi] * CC;
    {
      const float e0 = __expf(clip50(g00[r] + bg0));
      const float e1 = __expf(clip50(g01[r] + bg1));
      const float w0 = e0 / denom[sLo + t0 + lo];
      const float w1 = e1 / denom[sLo + t1 + lo];
      unsafeAtomicAdd(&numer[sLo + t0 + lo], (f00[r] + bf0) * w0);
      unsafeAtomicAdd(&numer[sLo + t1 + lo], (f01[r] + bf1) * w1);
    }
    {
      const float e0 = __expf(clip50(g10[r] + bg0));
      const float e1 = __expf(clip50(g11[r] + bg1));
      const float w0 = e0 / denom[sHi + t0 + lo];
      const float w1 = e1 / denom[sHi + t1 + lo];
      unsafeAtomicAdd(&numer[sHi + t0 + lo], (f10[r] + bf0) * w0);
      unsafeAtomicAdd(&numer[sHi + t1 + lo], (f11[r] + bf1) * w1);
    }
  }
}

// ----------------------------------------------- pass 3: y = numer @ Wh^T + bh
__global__ __launch_bounds__(256) void ygemm_kernel(
    const float* __restrict__ numer, const v2f* __restrict__ WhP,
    const float* __restrict__ bh, float* __restrict__ y) {
  __shared__ __align__(16) float sa[32][260];
  const int tid = threadIdx.x;
  const size_t rowbase = (size_t)blockIdx.x * 32;
  const float* as = numer + rowbase * CC;
#pragma unroll
  for (int it = 0; it < 8; ++it) {
    const int i = tid + it * 256;
    const int r = i >> 6, c4 = (i & 63) << 2;
    async_copy16(as + r * CC + c4, &sa[r][c4]);
  }
  async_wait0();
  __syncthreads();

  const int wave = tid >> 5, lane = tid & 31;
  const int half = lane >> 4, lo = lane & 15;
  const int t0 = wave * 32, t1 = t0 + 16;
  v8f h00 = {}, h01 = {}, h10 = {}, h11 = {};
  for (int k0 = 0; k0 < CC; k0 += 4) {
    const int kk = k0 + 2 * half;
    const v2f aLo = *(const v2f*)&sa[lo][kk];
    const v2f aHi = *(const v2f*)&sa[16 + lo][kk];
    const int kp = (k0 >> 1) + half;
    const v2f b0 = WhP[kp * CC + t0 + lo];
    const v2f b1 = WhP[kp * CC + t1 + lo];
    h00 = wmma_f32_k4(aLo, b0, h00);
    h01 = wmma_f32_k4(aLo, b1, h01);
    h10 = wmma_f32_k4(aHi, b0, h10);
    h11 = wmma_f32_k4(aHi, b1, h11);
  }
  const float bh0 = bh[t0 + lo], bh1 = bh[t1 + lo];
#pragma unroll
  for (int r = 0; r < 8; ++r) {
    const int mLo = r + 8 * half, mHi = mLo + 16;
    y[(rowbase + mLo) * CC + t0 + lo] = h00[r] + bh0;
    y[(rowbase + mLo) * CC + t1 + lo] = h01[r] + bh1;
    y[(rowbase + mHi) * CC + t0 + lo] = h10[r] + bh0;
    y[(rowbase + mHi) * CC + t1 + lo] = h11[r] + bh1;
  }
}

// ---------------------------------------------------- pass 4: out[n] = y[ix[n]]
__global__ __launch_bounds__(256) void gather_kernel(
    const float* __restrict__ y, const int* __restrict__ ix,
    float4* __restrict__ out) {
  const int i = blockIdx.x * 256 + threadIdx.x;  // one float4 per thread, NN*64 total
  const int n = i >> 6, c4 = i & 63;
  const int seg = ix[n];
  out[(size_t)n * 64 + c4] = ((const float4*)y)[(size_t)seg * 64 + c4];
}

// ---------------------------------------------------------------------------
extern "C" void kernel_launch(void* const* d_in, const int* in_sizes, int n_in,
                              void* d_out, int out_size, void* d_ws, size_t ws_size,
                              hipStream_t stream) {
  const float* x  = (const float*)d_in[0];
  const float* Wf = (const float*)d_in[1];
  const float* bf = (const float*)d_in[2];
  const float* Wg = (const float*)d_in[3];
  const float* bg = (const float*)d_in[4];
  const float* Wh = (const float*)d_in[5];
  const float* bh = (const float*)d_in[6];
  const int*   ix = (const int*)d_in[7];
  float* out = (float*)d_out;

  // Workspace layout (floats): WfP | WgP | WhP | denom | numer | y  (~13.4 MB)
  float* ws    = (float*)d_ws;
  v2f*   WfP   = (v2f*)ws;
  v2f*   WgP   = (v2f*)(ws + CC * CC);
  v2f*   WhP   = (v2f*)(ws + 2 * CC * CC);
  float* denom = ws + 3 * CC * CC;
  float* numer = denom + (size_t)GG * CC;
  float* y     = numer + (size_t)GG * CC;

  // 1) zero denom+numer (contiguous 2*GG*CC floats)
  zero_ws_kernel<<<(2 * GG * CC / 4) / 256, 256, 0, stream>>>(
      (float4*)denom, 2 * GG * CC / 4);
  // 2) pack the three weight matrices into paired-row transposed form
  pack_w_kernel<<<dim3(CC, 3), 128, 0, stream>>>(Wf, Wg, Wh, WfP, WgP, WhP);
  // 3) denom = segment_sum(exp(clip(x@Wg^T + bg)))
  pass1_denom_kernel<<<NN / 32, 256, 0, stream>>>(x, WgP, bg, ix, denom);
  // 4) numer = segment_sum((x@Wf^T + bf) * exp(...)/denom[seg])
  pass2_numer_kernel<<<NN / 32, 256, 0, stream>>>(x, WfP, bf, WgP, bg, ix, denom, numer);
  // 5) y = numer @ Wh^T + bh
  ygemm_kernel<<<GG / 32, 256, 0, stream>>>(numer, WhP, bh, y);
  // 6) out = y[ix]
  gather_kernel<<<(NN * 64) / 256, 256, 0, stream>>>(y, ix, (float4*)out);
}